// ClosedFormLoss_43920335569118
// MI455X (gfx1250) — compile-verified
//
#include <hip/hip_runtime.h>

// ---------------- problem constants ----------------
#define HH     513
#define WWD    513
#define HWP    (HH*WWD)        // 263169 pixels
#define WN     511             // windows per dim
#define WNW    (WN*WN)         // 261121 windows
#define NC     7
#define EPSR   1e-7f
#define TRI_CONF 100.0f

// ---------------- workspace layout (bytes) ----------------
// keepf:   WNW floats                    @ 0          (~1.0 MB)
// f4:      NC*WNW float4 {C,Qx,Qy,Qz}    @ 1 MB       (~27.9 MB)
// partial: 2*1028 floats                 @ ~30.3 MB   (all L2-resident)
#define OFF_KEEP 0u
#define OFF_F4   (1u<<20)
#define OFF_PART (OFF_F4 + 29245696u)   // 7*261121*16 rounded to 256
#define NBLK2    1028                    // floor(263169/256); last pixel gets no thread

typedef float v2f __attribute__((ext_vector_type(2)));
typedef float v8f __attribute__((ext_vector_type(8)));

// =====================================================================
// Pass 1: one wave per 3x3 window.
// D(16x16) += A(16x4)xB(4x16) three times (K = 9 padded to 12):
//   A rows 0..2 = r,g,b at window pixels, row 3 = 1  (P = [winI^T; 1^T])
//   B cols 0..6 = the 7 class probabilities at window pixels
// => lanes 0..6 end with (winI^T o, s) for class = lane, then each of
//    those lanes does the 3x3 regularized solve and stores float4 {C,Q}.
//
// Fragment setup is fully branch-free: masks are folded into FMAs so the
// loads are unconditionally consumed (no predication possible), and all
// addressing is 32-bit (saddr + scaled voffset).
// =====================================================================
__global__ __launch_bounds__(256)
void cf_pass1(const float* __restrict__ img,    // (H,W,3) this batch
              const float* __restrict__ cprob,  // (NC,H,W) this batch
              const int*   __restrict__ trimap, // (H,W) this batch
              float4* __restrict__ wsF4,        // (NC, WNW)
              float*  __restrict__ wsKeep)      // (WNW)
{
    const int lane = threadIdx.x & 31;
    const int wid  = blockIdx.x * 8 + (threadIdx.x >> 5);
    if (wid >= WNW) return;                 // wave-uniform: EXEC stays full
    const int wr = wid / WN, wc = wid % WN;
    const bool hi = (lane >> 4) != 0;       // half of the wave
    const int M = lane & 15;
    const float inv255 = 1.0f / 255.0f;

    // per-lane constants; loads below are always executed and consumed
    const int   ch  = (M < 3)  ? M : 0;           // clamped channel
    const int   cls = (M < NC) ? M : 0;           // clamped class
    const float mA0 = (M < 3)  ? inv255 : 0.0f;   // A scale mask
    const float oA0 = (M == 3) ? 1.0f   : 0.0f;   // A ones-row
    const float mB0 = (M < NC) ? 1.0f   : 0.0f;   // B mask

    // prefetch next window-row's image data (emits global_prefetch_b8)
    {
        int prow = (wr + 8 < HH) ? wr + 8 : wr;
        __builtin_prefetch(img + (prow * WWD + wc) * 3, 0, 3);
    }

    v8f acc = {0.f,0.f,0.f,0.f,0.f,0.f,0.f,0.f};
    #pragma unroll
    for (int t = 0; t < 3; ++t) {
        v2f a, b;
        #pragma unroll
        for (int j = 0; j < 2; ++j) {
            const int k0 = 4*t + j;         // K index when half==0
            const int k1 = 4*t + 2 + j;     // K index when half==1
            // compile-time row/col (clamped to pixel (2,2) when k>=9)
            const int r0 = (k0 < 9) ? k0/3 : 2, c0 = (k0 < 9) ? k0%3 : 2;
            const int r1 = (k1 < 9) ? k1/3 : 2, c1 = (k1 < 9) ? k1%3 : 2;
            const bool v0 = (k0 < 9), v1 = (k1 < 9);
            const int  kr = hi ? r1 : r0;               // cndmask of constants
            const int  kc = hi ? c1 : c0;
            const bool valid = hi ? v1 : v0;
            const float mA = valid ? mA0 : 0.0f;
            const float oA = valid ? oA0 : 0.0f;
            const float mB = valid ? mB0 : 0.0f;
            const int p = (wr + kr) * WWD + (wc + kc);  // 32-bit addressing
            // unconditional loads; masks folded into FMA/MUL
            a[j] = img[p*3 + ch] * mA + oA;
            b[j] = cprob[cls * HWP + p] * mB;
        }
        acc = __builtin_amdgcn_wmma_f32_16x16x4_f32(
                  false, a, false, b, (short)0, acc, false, false);
    }

    // finalize on lanes 0..6 (class = M); D layout: VGPR v, half 0 -> M-row v
    if (!hi && M < NC) {
        // keep flag: 5x5 OR of (trimap==128) centered at window center (uniform)
        int r0 = wr-1 < 0 ? 0 : wr-1, r1 = wr+3 > HH-1 ? HH-1 : wr+3;
        int c0 = wc-1 < 0 ? 0 : wc-1, c1 = wc+3 > WWD-1 ? WWD-1 : wc+3;
        int kp = 0;
        for (int r = r0; r <= r1; ++r)
            for (int c = c0; c <= c1; ++c)
                kp |= (trimap[r*WWD + c] == 128);
        const float keepf = kp ? 1.0f : 0.0f;
        if (M == 0) wsKeep[wid] = keepf;

        // mean / covariance / regularized inverse (uniform across the 7 lanes)
        float sx=0,sy=0,sz=0,xx=0,xy=0,xz=0,yy=0,yz=0,zz=0;
        #pragma unroll
        for (int k = 0; k < 9; ++k) {
            const int p = (wr + k/3) * WWD + (wc + k%3);
            const float r = img[p*3+0]*inv255, g = img[p*3+1]*inv255,
                        bl = img[p*3+2]*inv255;
            sx+=r; sy+=g; sz+=bl;
            xx+=r*r; xy+=r*g; xz+=r*bl; yy+=g*g; yz+=g*bl; zz+=bl*bl;
        }
        const float inv9 = 1.0f/9.0f, e = EPSR*inv9;
        const float mx=sx*inv9, my=sy*inv9, mz=sz*inv9;
        const float a00=xx*inv9-mx*mx+e, a01=xy*inv9-mx*my, a02=xz*inv9-mx*mz;
        const float a11=yy*inv9-my*my+e, a12=yz*inv9-my*mz, a22=zz*inv9-mz*mz+e;
        const float A00=a11*a22-a12*a12, A01=a02*a12-a01*a22, A02=a01*a12-a02*a11;
        const float A11=a00*a22-a02*a02, A12=a01*a02-a00*a12, A22=a00*a11-a01*a01;
        const float id = 1.0f/(a00*A00 + a01*A01 + a02*A02);
        const float i00=A00*id, i01=A01*id, i02=A02*id,
                    i11=A11*id, i12=A12*id, i22=A22*id;

        // WMMA result for class M: t' = winI^T o (acc[0..2]), s = sum(o) (acc[3])
        const float s = acc[3];
        const float t3x = acc[0]-mx*s, t3y = acc[1]-my*s, t3z = acc[2]-mz*s;
        const float vx = i00*t3x + i01*t3y + i02*t3z;
        const float vy = i01*t3x + i11*t3y + i12*t3z;
        const float vz = i02*t3x + i12*t3y + i22*t3z;
        float4 out;
        out.x = keepf * (s - (mx*vx + my*vy + mz*vz));  // C term
        out.y = keepf * vx; out.z = keepf * vy; out.w = keepf * vz;  // Q
        wsF4[M * WNW + wid] = out;
    }
}

// =====================================================================
// Pass 2: one thread per pixel; gather the <=9 windows containing it:
//   Lo = K*o - (C + img_p . Q)/9 ;  Ao = Lo + conf*o ;  sum (Ao-target)^2
// Deterministic block reduction -> per-block partial.
// =====================================================================
__global__ __launch_bounds__(256)
void cf_pass2(const float* __restrict__ img,
              const float* __restrict__ cprob,
              const int*   __restrict__ trimap,
              const float4* __restrict__ wsF4,
              const float*  __restrict__ wsKeep,
              float* __restrict__ partial)
{
    const int p = blockIdx.x * blockDim.x + threadIdx.x;
    float acc = 0.0f;
    if (p < HWP) {
        const int i = p / WWD, j = p % WWD;
        const int tri = trimap[p];
        const float conf = (tri != 128) ? TRI_CONF : 0.0f;
        const float inv255 = 1.0f/255.0f;
        const float ir = img[p*3+0]*inv255, ig = img[p*3+1]*inv255,
                    ib = img[p*3+2]*inv255;
        __builtin_prefetch(cprob + p, 0, 3);

        const int wr0 = i-2 < 0 ? 0 : i-2, wr1 = i > WN-1 ? WN-1 : i;
        const int wc0 = j-2 < 0 ? 0 : j-2, wc1 = j > WN-1 ? WN-1 : j;
        float K = 0.0f;
        float Cc[NC], Qx[NC], Qy[NC], Qz[NC];
        #pragma unroll
        for (int c = 0; c < NC; ++c) { Cc[c]=0.f; Qx[c]=0.f; Qy[c]=0.f; Qz[c]=0.f; }
        for (int wr = wr0; wr <= wr1; ++wr)
            for (int wc = wc0; wc <= wc1; ++wc) {
                const int w = wr*WN + wc;
                K += wsKeep[w];
                #pragma unroll
                for (int c = 0; c < NC; ++c) {
                    const float4 f = wsF4[c * WNW + w];
                    Cc[c] += f.x; Qx[c] += f.y; Qy[c] += f.z; Qz[c] += f.w;
                }
            }
        const float inv9 = 1.0f/9.0f;
        #pragma unroll
        for (int c = 0; c < NC; ++c) {
            const float o = cprob[c * HWP + p];
            const float Lo = K*o - (Cc[c] + ir*Qx[c] + ig*Qy[c] + ib*Qz[c])*inv9;
            const float target = (tri == c+1) ? TRI_CONF : 0.0f;
            const float d = Lo + conf*o - target;
            acc += d*d;
        }
    }
    __shared__ float red[256];
    red[threadIdx.x] = acc;
    __syncthreads();
    for (int s = 128; s > 0; s >>= 1) {
        if ((int)threadIdx.x < s) red[threadIdx.x] += red[threadIdx.x + s];
        __syncthreads();
    }
    if (threadIdx.x == 0) partial[blockIdx.x] = red[0];
}

// =====================================================================
// Pass 3: fixed-order final reduction, scale by 1/n^2 (n = H*W).
// =====================================================================
__global__ void cf_pass3(const float* __restrict__ partial, int n,
                         float* __restrict__ out)
{
    __shared__ float red[256];
    float a = 0.0f;
    for (int i = threadIdx.x; i < n; i += 256) a += partial[i];
    red[threadIdx.x] = a;
    __syncthreads();
    for (int s = 128; s > 0; s >>= 1) {
        if ((int)threadIdx.x < s) red[threadIdx.x] += red[threadIdx.x + s];
        __syncthreads();
    }
    if (threadIdx.x == 0) {
        const float invn = 1.0f / (float)HWP;
        out[0] = red[0] * invn * invn;
    }
}

// =====================================================================
extern "C" void kernel_launch(void* const* d_in, const int* in_sizes, int n_in,
                              void* d_out, int out_size, void* d_ws, size_t ws_size,
                              hipStream_t stream)
{
    const float* cprob  = (const float*)d_in[0];  // (2, 7, 513, 513)
    const float* img    = (const float*)d_in[1];  // (2, 513, 513, 3)
    const int*   trimap = (const int*)  d_in[2];  // (2, 513, 513)

    char*   ws     = (char*)d_ws;
    float*  wsKeep = (float*) (ws + OFF_KEEP);
    float4* wsF4   = (float4*)(ws + OFF_F4);
    float*  wsPart = (float*) (ws + OFF_PART);

    const int g1 = (WNW + 7) / 8;          // 8 waves (windows) per 256-thread block
    const int g2 = NBLK2;

    for (int b = 0; b < 2; ++b) {
        const float* imgB = img    + (size_t)b * HWP * 3;
        const float* cpB  = cprob  + (size_t)b * NC * HWP;
        const int*   trB  = trimap + (size_t)b * HWP;
        cf_pass1<<<g1, 256, 0, stream>>>(imgB, cpB, trB, wsF4, wsKeep);
        cf_pass2<<<g2, 256, 0, stream>>>(imgB, cpB, trB, wsF4, wsKeep,
                                         wsPart + (size_t)b * g2);
    }
    cf_pass3<<<1, 256, 0, stream>>>(wsPart, 2 * g2, (float*)d_out);
}